// GT_TGN_14181982011704
// MI455X (gfx1250) — compile-verified
//
#include <hip/hip_runtime.h>
#include <math.h>

typedef float v2f __attribute__((ext_vector_type(2)));
typedef float v8f __attribute__((ext_vector_type(8)));

#define N_NODES 4096
#define E_EDGES 32768
#define C_CAND  64
#define IN_DIM  128
#define H_DIM   256
#define QKV_LD  768            // 3*H
#define HEADS   4
#define HD      64
#define CAP     64             // per-node edge-list capacity (avg degree 8)

// GEMM blocking
#define TM   64
#define TN   64
#define TK   32
#define TKP  36                // padded LDS row: 144B (16B aligned rows, conflict-free b64)

// -------- CDNA5 async global->LDS copy (ASYNCcnt path), with safe fallback ----
#if defined(__has_builtin)
#if __has_builtin(__builtin_amdgcn_global_load_async_to_lds_b128) && \
    __has_builtin(__builtin_amdgcn_s_wait_asynccnt)
#define USE_ASYNC 1
#endif
#endif

#ifdef USE_ASYNC
// builtin param type per hipcc diagnostic: 'int __vector(4) __device__ *'
typedef __attribute__((__vector_size__(4 * sizeof(int)))) int ai4;
typedef __attribute__((address_space(1))) ai4* as1_ai4_p;
typedef __attribute__((address_space(3))) ai4* as3_ai4_p;
#endif

__device__ __forceinline__ void async_cp16(const float* g, float* l) {
#ifdef USE_ASYNC
    __builtin_amdgcn_global_load_async_to_lds_b128((as1_ai4_p)g, (as3_ai4_p)l, 0, 0);
#else
    *(float4*)l = *(const float4*)g;
#endif
}

__device__ __forceinline__ void async_wait_all() {
#ifdef USE_ASYNC
    __builtin_amdgcn_s_wait_asynccnt(0);
#endif
}

__device__ __forceinline__ v8f wmma_f32(v2f a, v2f b, v8f c) {
    // D = A(16x4 f32) * B(4x16 f32) + C(16x16 f32)
    return __builtin_amdgcn_wmma_f32_16x16x4_f32(false, a, false, b, (short)0, c,
                                                 false, false);
}

__device__ __forceinline__ float sigmoidf_(float x) { return 1.0f / (1.0f + __expf(-x)); }

// ---------------------------------------------------------------------------
// WMMA NT GEMM: C[M,Nn] = A[M,K] @ B[Nn,K]^T + bias[Nn] (+res) (relu?)
// 128-thread block (4 waves) computes a 64x64 tile; wave w owns rows 16w..16w+15.
// K blocked by 32, A/B tiles double-buffered in LDS, filled with async copies.
// grid = (Nn/64, M/64), block = 128. Requires M%64==0, Nn%64==0, K%32==0.
// ---------------------------------------------------------------------------
__global__ void gemm_nt_wmma(const float* __restrict__ A, const float* __restrict__ B,
                             const float* __restrict__ bias,
                             const float* __restrict__ residual,
                             float* __restrict__ C, int M, int Nn, int K, int relu) {
    __shared__ float As[2][TM][TKP];
    __shared__ float Bs[2][TN][TKP];

    const int tid  = threadIdx.x;
    const int w    = tid >> 5;
    const int lane = tid & 31;
    const int r = lane & 15;
    const int h = lane >> 4;
    const int m0 = blockIdx.y << 6;
    const int n0 = blockIdx.x << 6;

    // fill mapping: thread -> (row, 16-col segment); 4x b128 per tile per thread
    const int frow = tid >> 1;
    const int fseg = (tid & 1) << 4;
    const float* Ag = A + (size_t)(m0 + frow) * K + fseg;
    const float* Bg = B + (size_t)(n0 + frow) * K + fseg;

    v8f acc0 = {0.f,0.f,0.f,0.f,0.f,0.f,0.f,0.f};
    v8f acc1 = acc0, acc2 = acc0, acc3 = acc0;

    const int nK = K / TK;

    // prologue: fill buffer 0
#pragma unroll
    for (int j = 0; j < 16; j += 4) {
        async_cp16(Ag + j, &As[0][frow][fseg + j]);
        async_cp16(Bg + j, &Bs[0][frow][fseg + j]);
    }
    async_wait_all();
    __syncthreads();

    for (int kb = 0; kb < nK; ++kb) {
        const int cur = kb & 1, nxt = cur ^ 1;
        if (kb + 1 < nK) {
            const float* Agn = Ag + (size_t)(kb + 1) * TK;
            const float* Bgn = Bg + (size_t)(kb + 1) * TK;
#pragma unroll
            for (int j = 0; j < 16; j += 4) {
                async_cp16(Agn + j, &As[nxt][frow][fseg + j]);
                async_cp16(Bgn + j, &Bs[nxt][frow][fseg + j]);
            }
        }
        // compute on current buffer: 8 K-steps of 4
#pragma unroll
        for (int s = 0; s < 8; ++s) {
            v2f a  = *(const v2f*)&As[cur][(w << 4) + r][4 * s + 2 * h];
            v2f b0 = *(const v2f*)&Bs[cur][r][4 * s + 2 * h];
            v2f b1 = *(const v2f*)&Bs[cur][16 + r][4 * s + 2 * h];
            v2f b2 = *(const v2f*)&Bs[cur][32 + r][4 * s + 2 * h];
            v2f b3 = *(const v2f*)&Bs[cur][48 + r][4 * s + 2 * h];
            acc0 = wmma_f32(a, b0, acc0);
            acc1 = wmma_f32(a, b1, acc1);
            acc2 = wmma_f32(a, b2, acc2);
            acc3 = wmma_f32(a, b3, acc3);
        }
        async_wait_all();     // next buffer resident
        __syncthreads();      // all waves done reading current buffer
    }

    const float bb0 = bias[n0 + r];
    const float bb1 = bias[n0 + 16 + r];
    const float bb2 = bias[n0 + 32 + r];
    const float bb3 = bias[n0 + 48 + r];
#pragma unroll
    for (int i = 0; i < 8; ++i) {
        size_t row  = (size_t)(m0 + (w << 4) + i + 8 * h);
        size_t base = row * (size_t)Nn + n0 + r;
        float v0 = acc0[i] + bb0;
        float v1 = acc1[i] + bb1;
        float v2 = acc2[i] + bb2;
        float v3 = acc3[i] + bb3;
        if (residual) {
            v0 += residual[base];
            v1 += residual[base + 16];
            v2 += residual[base + 32];
            v3 += residual[base + 48];
        }
        if (relu) {
            v0 = fmaxf(v0, 0.f); v1 = fmaxf(v1, 0.f);
            v2 = fmaxf(v2, 0.f); v3 = fmaxf(v3, 0.f);
        }
        C[base]      = v0;
        C[base + 16] = v1;
        C[base + 32] = v2;
        C[base + 48] = v3;
    }
}

// ---------------------------------------------------------------------------
// Flash attention, one wave per (16 query rows, 1 head). grid=(N/16, HEADS).
// K-fragments batched into registers before the WMMA chain; V fragments
// prefetched before the softmax math so loads overlap exp/shuffles.
// ---------------------------------------------------------------------------
__global__ void flash_attn(const float* __restrict__ qkv, float* __restrict__ out) {
    const int lane = threadIdx.x & 31;
    const int r = lane & 15;
    const int h = lane >> 4;
    const int q0 = blockIdx.x << 4;
    const int g  = blockIdx.y;

    __shared__ float Ps[16][16];

    v2f qa[16];
    const float* qrow = qkv + (size_t)(q0 + r) * QKV_LD + g * HD + 2 * h;
#pragma unroll
    for (int kk = 0; kk < 16; ++kk) qa[kk] = *(const v2f*)(qrow + kk * 4);

    v8f o0 = {0.f,0.f,0.f,0.f,0.f,0.f,0.f,0.f};
    v8f o1 = o0, o2 = o0, o3 = o0;
    float mrun[8], lrun[8];
#pragma unroll
    for (int i = 0; i < 8; ++i) { mrun[i] = -INFINITY; lrun[i] = 0.f; }

    const float scale = 0.125f; // 1/sqrt(64)

    for (int kt = 0; kt < N_NODES / 16; ++kt) {
        // batch-load all 16 K fragments, then run the S WMMA chain
        const float* krow = qkv + (size_t)(kt * 16 + r) * QKV_LD + H_DIM + g * HD + 2 * h;
        v2f kb[16];
#pragma unroll
        for (int kk = 0; kk < 16; ++kk) kb[kk] = *(const v2f*)(krow + kk * 4);
        v8f s = {0.f,0.f,0.f,0.f,0.f,0.f,0.f,0.f};
#pragma unroll
        for (int kk = 0; kk < 16; ++kk) s = wmma_f32(qa[kk], kb[kk], s);

        // prefetch V fragments (independent of softmax)
        const float* vbase = qkv + (size_t)(kt * 16) * QKV_LD + 2 * H_DIM + g * HD;
        v2f vb0[4], vb1[4], vb2[4], vb3[4];
#pragma unroll
        for (int kk = 0; kk < 4; ++kk) {
            const float* v0p = vbase + (size_t)(kk * 4 + 2 * h) * QKV_LD;
            const float* v1p = v0p + QKV_LD;
            vb0[kk] = v2f{ v0p[r],      v1p[r]      };
            vb1[kk] = v2f{ v0p[16 + r], v1p[16 + r] };
            vb2[kk] = v2f{ v0p[32 + r], v1p[32 + r] };
            vb3[kk] = v2f{ v0p[48 + r], v1p[48 + r] };
        }

        // online softmax; row m = i + 8h lives in the lane's 16-lane half
        float alpha[8];
#pragma unroll
        for (int i = 0; i < 8; ++i) {
            float sv = s[i] * scale;
            float t = sv;
            t = fmaxf(t, __shfl_xor(t, 1, 32));
            t = fmaxf(t, __shfl_xor(t, 2, 32));
            t = fmaxf(t, __shfl_xor(t, 4, 32));
            t = fmaxf(t, __shfl_xor(t, 8, 32));
            float mnew = fmaxf(mrun[i], t);
            float p  = __expf(sv - mnew);
            float rs = p;
            rs += __shfl_xor(rs, 1, 32);
            rs += __shfl_xor(rs, 2, 32);
            rs += __shfl_xor(rs, 4, 32);
            rs += __shfl_xor(rs, 8, 32);
            alpha[i] = __expf(mrun[i] - mnew);
            lrun[i]  = lrun[i] * alpha[i] + rs;
            mrun[i]  = mnew;
            Ps[i + 8 * h][r] = p;   // C-layout -> row-major P (transpose to A-layout)
        }
        __syncthreads();

#pragma unroll
        for (int i = 0; i < 8; ++i) {
            o0[i] *= alpha[i]; o1[i] *= alpha[i];
            o2[i] *= alpha[i]; o3[i] *= alpha[i];
        }

        // O += P(16x16) @ V(16x64)
#pragma unroll
        for (int kk = 0; kk < 4; ++kk) {
            v2f p = *(const v2f*)&Ps[r][kk * 4 + 2 * h];
            o0 = wmma_f32(p, vb0[kk], o0);
            o1 = wmma_f32(p, vb1[kk], o1);
            o2 = wmma_f32(p, vb2[kk], o2);
            o3 = wmma_f32(p, vb3[kk], o3);
        }
        __syncthreads();
    }

#pragma unroll
    for (int i = 0; i < 8; ++i) {
        size_t row = (size_t)(q0 + i + 8 * h);
        float inv = 1.0f / lrun[i];
        float* op = out + row * H_DIM + g * HD + r;
        op[0]  = o0[i] * inv;
        op[16] = o1[i] * inv;
        op[32] = o2[i] * inv;
        op[48] = o3[i] * inv;
    }
}

// ---------------------------------------------------------------------------
// Row LayerNorm over H=256. grid = N rows, block = 256.
// ---------------------------------------------------------------------------
__global__ void layernorm_k(const float* __restrict__ in, const float* __restrict__ g,
                            const float* __restrict__ b, float* __restrict__ out) {
    const int row = blockIdx.x, t = threadIdx.x;
    __shared__ float red[256];
    float v = in[(size_t)row * H_DIM + t];
    red[t] = v; __syncthreads();
    for (int s = 128; s > 0; s >>= 1) {
        if (t < s) red[t] += red[t + s];
        __syncthreads();
    }
    float mean = red[0] * (1.0f / H_DIM);
    __syncthreads();
    float d = v - mean;
    red[t] = d * d; __syncthreads();
    for (int s = 128; s > 0; s >>= 1) {
        if (t < s) red[t] += red[t + s];
        __syncthreads();
    }
    float var = red[0] * (1.0f / H_DIM);
    out[(size_t)row * H_DIM + t] = g[t] * d * rsqrtf(var + 1e-5f) + b[t];
}

// ---------------------------------------------------------------------------
// Stable per-destination edge lists: one thread per dst scans all edges in
// order (preserves the sequential-scan semantics within each dst chain).
// ---------------------------------------------------------------------------
__global__ void build_lists(const int* __restrict__ edges, int* __restrict__ lists,
                            int* __restrict__ counts) {
    int d = blockIdx.x * blockDim.x + threadIdx.x;
    if (d >= N_NODES) return;
    int cnt = 0;
    for (int e = 0; e < E_EDGES; ++e) {
        if (edges[2 * e + 1] == d && cnt < CAP) lists[d * CAP + cnt++] = e;
    }
    counts[d] = cnt;
}

// ---------------------------------------------------------------------------
// Per-node GRU chain. grid = N nodes, block = 256 (h state in LDS).
// gi = nf_proj (node_features @ w_ih^T + b_ih), indexed by edge src.
// ---------------------------------------------------------------------------
__global__ void gru_scan(const float* __restrict__ gi, const int* __restrict__ edges,
                         const int* __restrict__ lists, const int* __restrict__ counts,
                         const float* __restrict__ w_hh, const float* __restrict__ b_hh,
                         float* __restrict__ mem) {
    const int d = blockIdx.x, t = threadIdx.x;
    __shared__ float hsh[H_DIM];
    hsh[t] = 0.f;
    __syncthreads();
    const int cnt = counts[d];
    const float* wr = w_hh + (size_t)t * H_DIM;
    const float* wz = w_hh + (size_t)(H_DIM + t) * H_DIM;
    const float* wn = w_hh + (size_t)(2 * H_DIM + t) * H_DIM;
    for (int j = 0; j < cnt; ++j) {
        int e   = lists[d * CAP + j];
        int src = edges[2 * e];
        const float* girow = gi + (size_t)src * QKV_LD;
        float ghr = b_hh[t], ghz = b_hh[H_DIM + t], ghn = b_hh[2 * H_DIM + t];
        for (int k = 0; k < H_DIM; ++k) {
            float hk = hsh[k];
            ghr = fmaf(wr[k], hk, ghr);
            ghz = fmaf(wz[k], hk, ghz);
            ghn = fmaf(wn[k], hk, ghn);
        }
        float rg = sigmoidf_(girow[t] + ghr);
        float zg = sigmoidf_(girow[H_DIM + t] + ghz);
        float ng = tanhf(girow[2 * H_DIM + t] + rg * ghn);
        float hnew = (1.0f - zg) * ng + zg * hsh[t];
        __syncthreads();          // all reads of hsh done
        hsh[t] = hnew;
        __syncthreads();
    }
    mem[(size_t)d * H_DIM + t] = hsh[t];
}

// ---------------------------------------------------------------------------
// Candidate gather: hcat[c] = concat(refined[cand[c][0]], refined[cand[c][1]])
// ---------------------------------------------------------------------------
__global__ void gather_hcat(const float* __restrict__ refined, const int* __restrict__ cand,
                            float* __restrict__ hcat) {
    int c = blockIdx.x, k = threadIdx.x; // block = 512
    int node = cand[c * 2 + (k >> 8)];
    hcat[(size_t)c * 512 + k] = refined[(size_t)node * H_DIM + (k & 255)];
}

// ---------------------------------------------------------------------------
// Final Linear(H,1) + sigmoid. grid = C, block = 32.
// ---------------------------------------------------------------------------
__global__ void ep2_sigmoid(const float* __restrict__ eph, const float* __restrict__ w2,
                            const float* __restrict__ b2, float* __restrict__ out) {
    int c = blockIdx.x, lane = threadIdx.x;
    float s = 0.f;
    for (int k = lane; k < H_DIM; k += 32) s = fmaf(eph[(size_t)c * H_DIM + k], w2[k], s);
    s += __shfl_xor(s, 1, 32);
    s += __shfl_xor(s, 2, 32);
    s += __shfl_xor(s, 4, 32);
    s += __shfl_xor(s, 8, 32);
    s += __shfl_xor(s, 16, 32);
    if (lane == 0) out[c] = sigmoidf_(s + b2[0]);
}

// ---------------------------------------------------------------------------
extern "C" void kernel_launch(void* const* d_in, const int* in_sizes, int n_in,
                              void* d_out, int out_size, void* d_ws, size_t ws_size,
                              hipStream_t stream) {
    const float* node_features = (const float*)d_in[0];
    const int*   edges         = (const int*)  d_in[1];
    const int*   candidates    = (const int*)  d_in[2];
    const float* gru_w_ih      = (const float*)d_in[3];
    const float* gru_w_hh      = (const float*)d_in[4];
    const float* gru_b_ih      = (const float*)d_in[5];
    const float* gru_b_hh      = (const float*)d_in[6];
    const float* attn_in_w     = (const float*)d_in[7];
    const float* attn_in_b     = (const float*)d_in[8];
    const float* attn_out_w    = (const float*)d_in[9];
    const float* attn_out_b    = (const float*)d_in[10];
    const float* ln1_g         = (const float*)d_in[11];
    const float* ln1_b         = (const float*)d_in[12];
    const float* ffn_w1        = (const float*)d_in[13];
    const float* ffn_b1        = (const float*)d_in[14];
    const float* ffn_w2        = (const float*)d_in[15];
    const float* ffn_b2        = (const float*)d_in[16];
    const float* ln2_g         = (const float*)d_in[17];
    const float* ln2_b         = (const float*)d_in[18];
    const float* scn_w1        = (const float*)d_in[19];
    const float* scn_b1        = (const float*)d_in[20];
    const float* scn_w2        = (const float*)d_in[21];
    const float* scn_b2        = (const float*)d_in[22];
    const float* ep_w1         = (const float*)d_in[23];
    const float* ep_b1         = (const float*)d_in[24];
    const float* ep_w2         = (const float*)d_in[25];
    const float* ep_b2         = (const float*)d_in[26];
    float* out = (float*)d_out;

    // workspace carve-up (~48 MB total)
    char* ws = (char*)d_ws;
    size_t off = 0;
    auto carve = [&](size_t bytes) -> void* {
        void* p = (void*)(ws + off);
        off += (bytes + 255) & ~(size_t)255;
        return p;
    };
    float* bufA    = (float*)carve((size_t)N_NODES * QKV_LD * 4);   // nf_proj, then qkv
    float* f1      = (float*)carve((size_t)N_NODES * 1024 * 4);     // FFN hidden
    float* x       = (float*)carve((size_t)N_NODES * H_DIM * 4);    // mem -> running x
    float* attn_o  = (float*)carve((size_t)N_NODES * H_DIM * 4);
    float* tmp1    = (float*)carve((size_t)N_NODES * H_DIM * 4);
    float* refined = (float*)carve((size_t)N_NODES * H_DIM * 4);
    int*   lists   = (int*)  carve((size_t)N_NODES * CAP * 4);
    int*   counts  = (int*)  carve((size_t)N_NODES * 4);
    float* hcat    = (float*)carve((size_t)C_CAND * 2 * H_DIM * 4);
    float* eph     = (float*)carve((size_t)C_CAND * H_DIM * 4);
    (void)ws_size; (void)in_sizes; (void)n_in; (void)out_size;

    // 1) GRU input projection: nf_proj = node_features @ w_ih^T + b_ih  [4096,768]
    gemm_nt_wmma<<<dim3(QKV_LD / TN, N_NODES / TM), 128, 0, stream>>>(
        node_features, gru_w_ih, gru_b_ih, nullptr, bufA, N_NODES, QKV_LD, IN_DIM, 0);

    // 2) stable per-dst edge lists
    build_lists<<<dim3(N_NODES / 256), 256, 0, stream>>>(edges, lists, counts);

    // 3) per-node GRU chains -> x (= mem)
    gru_scan<<<dim3(N_NODES), 256, 0, stream>>>(bufA, edges, lists, counts,
                                                gru_w_hh, gru_b_hh, x);

    // 4) transformer layers
    for (int l = 0; l < 2; ++l) {
        const float* inw  = attn_in_w  + (size_t)l * QKV_LD * H_DIM;
        const float* inb  = attn_in_b  + (size_t)l * QKV_LD;
        const float* outw = attn_out_w + (size_t)l * H_DIM * H_DIM;
        const float* outb = attn_out_b + (size_t)l * H_DIM;
        const float* w1   = ffn_w1 + (size_t)l * 1024 * H_DIM;
        const float* b1   = ffn_b1 + (size_t)l * 1024;
        const float* w2   = ffn_w2 + (size_t)l * H_DIM * 1024;
        const float* b2   = ffn_b2 + (size_t)l * H_DIM;

        // qkv = x @ in_w^T + in_b
        gemm_nt_wmma<<<dim3(QKV_LD / TN, N_NODES / TM), 128, 0, stream>>>(
            x, inw, inb, nullptr, bufA, N_NODES, QKV_LD, H_DIM, 0);
        // streaming-softmax attention
        flash_attn<<<dim3(N_NODES / 16, HEADS), 32, 0, stream>>>(bufA, attn_o);
        // out-proj + residual, then LN1
        gemm_nt_wmma<<<dim3(H_DIM / TN, N_NODES / TM), 128, 0, stream>>>(
            attn_o, outw, outb, x, tmp1, N_NODES, H_DIM, H_DIM, 0);
        layernorm_k<<<dim3(N_NODES), 256, 0, stream>>>(tmp1, ln1_g + l * H_DIM,
                                                       ln1_b + l * H_DIM, x);
        // FFN: relu(x@w1^T+b1)@w2^T+b2 + residual, then LN2
        gemm_nt_wmma<<<dim3(1024 / TN, N_NODES / TM), 128, 0, stream>>>(
            x, w1, b1, nullptr, f1, N_NODES, 1024, H_DIM, 1);
        gemm_nt_wmma<<<dim3(H_DIM / TN, N_NODES / TM), 128, 0, stream>>>(
            f1, w2, b2, x, tmp1, N_NODES, H_DIM, 1024, 0);
        layernorm_k<<<dim3(N_NODES), 256, 0, stream>>>(tmp1, ln2_g + l * H_DIM,
                                                       ln2_b + l * H_DIM, x);
    }

    // 5) SCN refiner
    gemm_nt_wmma<<<dim3(H_DIM / TN, N_NODES / TM), 128, 0, stream>>>(
        x, scn_w1, scn_b1, nullptr, tmp1, N_NODES, H_DIM, H_DIM, 1);
    gemm_nt_wmma<<<dim3(H_DIM / TN, N_NODES / TM), 128, 0, stream>>>(
        tmp1, scn_w2, scn_b2, nullptr, refined, N_NODES, H_DIM, H_DIM, 0);

    // 6) edge predictor on candidates (M=64 -> one block row)
    gather_hcat<<<dim3(C_CAND), 512, 0, stream>>>(refined, candidates, hcat);
    gemm_nt_wmma<<<dim3(H_DIM / TN, C_CAND / TM), 128, 0, stream>>>(
        hcat, ep_w1, ep_b1, nullptr, eph, C_CAND, H_DIM, 2 * H_DIM, 1);
    ep2_sigmoid<<<dim3(C_CAND), 32, 0, stream>>>(eph, ep_w2, ep_b2, out);
}